// PointLossHungarianMatcher_24163486007425
// MI455X (gfx1250) — compile-verified
//
#include <hip/hip_runtime.h>
#include <stdint.h>

// ---------------------------------------------------------------------------
// DETR HungarianMatcher cost matrix on MI455X (gfx1250).
//   C[i,j] = L1(p_i, t_j) - GIoU(p_i, t_j),  N=8192 preds, M=4096 targets.
//
// Output-streaming kernel: 134 MB of f32 writes dominate (floor ~5.8 us at
// 23.3 TB/s); inner loop is ~35 VALU/element, so VALU and store bandwidth are
// roughly balanced. Strategy:
//   * Stage the block's 2048-target tile (32 KB) into LDS with CDNA5 async
//     global->LDS copies (ASYNCcnt path, drained by s_wait_asynccnt).
//   * Lane t handles columns {jb+t, jb+32+t, jb+64+t, jb+96+t}: every
//     ds_load_b128 then has a 16B inter-lane stride (4 banks) -> conflict-free
//     2-phase LDS reads (the naive 4t..4t+3 mapping is an 8-way conflict).
//   * Each of 8 waves walks 4 pred rows; pred box is wave-uniform (one
//     broadcast fetch). Stores are scalar non-temporal, 128 B contiguous per
//     wave per store — write-only stream, bypasses cache.
//   * Divides replaced by v_rcp_f32 * mul (~2 ulp; fine for a cost matrix).
// ---------------------------------------------------------------------------

typedef float f32x4 __attribute__((ext_vector_type(4)));  // native clang vector

#define EPS        1e-16f
#define TJ         2048          // target columns staged per block (32 KB LDS)
#define ROWS       32            // pred rows per block
#define TX         32            // lanes (wave32)
#define TY         8             // waves per block
#define GROUP      (TX * 4)      // 128 columns per sweep
#define NSWEEP     (TJ / GROUP)  // 16 sweeps per row

__device__ __forceinline__ float pair_cost(const f32x4 p, const float parea,
                                           const f32x4 t) {
  // target area (recomputed: 3 VALU, cheaper than extra LDS traffic)
  const float tarea = (t.z - t.x) * (t.w - t.y);
  // L1 distance: subs + abs-modified adds
  const float l1 = fabsf(p.x - t.x) + fabsf(p.y - t.y) +
                   fabsf(p.z - t.z) + fabsf(p.w - t.w);
  // intersection
  float iw = fminf(p.z, t.z) - fmaxf(p.x, t.x);
  float ih = fminf(p.w, t.w) - fmaxf(p.y, t.y);
  iw = fmaxf(iw, 0.0f);
  ih = fmaxf(ih, 0.0f);
  const float inter = iw * ih;
  const float uni   = parea + tarea - inter + EPS;
  // enclosing box
  float ew = fmaxf(p.z, t.z) - fminf(p.x, t.x);
  float eh = fmaxf(p.w, t.w) - fminf(p.y, t.y);
  ew = fmaxf(ew, 0.0f);
  eh = fmaxf(eh, 0.0f);
  const float earea = ew * eh + EPS;
  // iou - (earea - uni)/earea  ==  inter/uni + uni/earea - 1
  const float iou  = inter * __builtin_amdgcn_rcpf(uni);
  const float giou = iou + (uni - earea) * __builtin_amdgcn_rcpf(earea);
  // C = 1*l1 - 1*giou
  return l1 - giou;
}

__global__ __launch_bounds__(TX * TY)
void matcher_cost_kernel(const float* __restrict__ out_bbox,  // [N,4]
                         const float* __restrict__ tgt_bbox,  // [M,4]
                         float* __restrict__ C,               // [N,M]
                         int N, int M) {
  __shared__ f32x4 sT[TJ];  // staged target boxes (AoS, 16B each)

  const int j0    = blockIdx.x * TJ;
  const int i0    = blockIdx.y * ROWS;
  const int tid   = threadIdx.y * TX + threadIdx.x;
  const int ncols = min(TJ, M - j0);

  // ---- CDNA5 async global->LDS staging (no VGPR round-trip) --------------
  // Each active lane moves one 16B box per issue; 8 issues per thread.
  for (int k = tid; k < ncols; k += TX * TY) {
    // generic LDS pointer: low 32 bits are the LDS byte offset (aperture form)
    unsigned lds_addr = (unsigned)(uintptr_t)(&sT[k]);
    const float* gp = tgt_bbox + (size_t)(j0 + k) * 4;
    asm volatile("global_load_async_to_lds_b128 %0, %1, off"
                 :
                 : "v"(lds_addr), "v"(gp)
                 : "memory");
  }
  asm volatile("s_wait_asynccnt 0" ::: "memory");  // drain this wave's ASYNCcnt
  __syncthreads();                                 // make LDS visible block-wide

  const bool full_rows = (i0 + ROWS <= N);
  const bool full_cols = (ncols == TJ);

  // ---- Compute: 8 waves x 4 rows, 4 strided columns per thread per sweep --
  for (int r = 0; r < ROWS / TY; ++r) {
    const int i = i0 + threadIdx.y + TY * r;
    if (!full_rows && i >= N) continue;

    // wave-uniform pred box: all 32 lanes load the same 16B -> one fetch
    const f32x4 p = ((const f32x4*)out_bbox)[i];
    const float parea = (p.z - p.x) * (p.w - p.y);
    float* __restrict__ crow = C + (size_t)i * M + j0;

    if (full_cols) {
      // Fast path: no guards, conflict-free LDS reads, coalesced NT stores.
      for (int s = 0; s < NSWEEP; ++s) {
        const int c0 = s * GROUP + threadIdx.x;
        const f32x4 t0 = sT[c0 + 0 * TX];
        const f32x4 t1 = sT[c0 + 1 * TX];
        const f32x4 t2 = sT[c0 + 2 * TX];
        const f32x4 t3 = sT[c0 + 3 * TX];
        __builtin_nontemporal_store(pair_cost(p, parea, t0), crow + c0 + 0 * TX);
        __builtin_nontemporal_store(pair_cost(p, parea, t1), crow + c0 + 1 * TX);
        __builtin_nontemporal_store(pair_cost(p, parea, t2), crow + c0 + 2 * TX);
        __builtin_nontemporal_store(pair_cost(p, parea, t3), crow + c0 + 3 * TX);
      }
    } else {
      // Ragged tail (only if M % TJ != 0)
      for (int s = 0; s < NSWEEP; ++s) {
        const int c0 = s * GROUP + threadIdx.x;
        #pragma unroll
        for (int u = 0; u < 4; ++u) {
          const int c = c0 + u * TX;
          if (c < ncols) {
            __builtin_nontemporal_store(pair_cost(p, parea, sT[c]), crow + c);
          }
        }
      }
    }
  }
}

extern "C" void kernel_launch(void* const* d_in, const int* in_sizes, int n_in,
                              void* d_out, int out_size, void* d_ws, size_t ws_size,
                              hipStream_t stream) {
  const float* out_bbox = (const float*)d_in[0];  // [N,4] f32
  const float* tgt_bbox = (const float*)d_in[1];  // [M,4] f32
  float* C = (float*)d_out;                       // [N,M] f32

  const int N = in_sizes[0] / 4;
  const int M = in_sizes[1] / 4;

  dim3 block(TX, TY);
  dim3 grid((M + TJ - 1) / TJ, (N + ROWS - 1) / ROWS);
  matcher_cost_kernel<<<grid, block, 0, stream>>>(out_bbox, tgt_bbox, C, N, M);
}